// GCN_67731634258537
// MI455X (gfx1250) — compile-verified
//
#include <hip/hip_runtime.h>

#define N_NODES 50000
#define N_EDGES 800000
#define D_IN 96
#define D_HID 256
#define D_OUT 2
#define N_GRAPHS 64

typedef __attribute__((ext_vector_type(2))) float v2f;
typedef __attribute__((ext_vector_type(8))) float v8f;

// ---------------- elementwise helpers ----------------

__global__ void copy_f4_kernel(const float4* __restrict__ in,
                               float4* __restrict__ out, int n4) {
    int i = blockIdx.x * blockDim.x + threadIdx.x;
    if (i < n4) out[i] = in[i];
}

__global__ void zero_f_kernel(float* __restrict__ p, int n) {
    int i = blockIdx.x * blockDim.x + threadIdx.x;
    if (i < n) p[i] = 0.0f;
}

// ---------------- edge scatter-add: out[dst] += xin[src] ----------------
// One thread per (edge, 4-feature group). float4 gather, 4 scalar global atomics.

__global__ void scatter_add_kernel(const float* __restrict__ xin,
                                   const int* __restrict__ src,
                                   const int* __restrict__ dst,
                                   float* __restrict__ out,
                                   int fgroups, int F) {
    long long idx = (long long)blockIdx.x * blockDim.x + threadIdx.x;
    long long tot = (long long)N_EDGES * fgroups;
    if (idx >= tot) return;
    int e = (int)(idx / fgroups);
    int j = (int)(idx % fgroups);
    int s = src[e];
    int d = dst[e];
    float4 v = *(const float4*)(xin + (long long)s * F + 4 * j);
    float* o = out + (long long)d * F + 4 * j;
    atomicAdd(o + 0, v.x);
    atomicAdd(o + 1, v.y);
    atomicAdd(o + 2, v.z);
    atomicAdd(o + 3, v.w);
}

// ---------------- fp32 WMMA GEMM: OUT[N,256] = act(A[N,K] @ W[K,256] + bias) ----
// Block = 256 threads = 8 waves; block owns one 16-row M-tile, stages it in LDS.
// Each wave computes two 16x16 N-tiles with V_WMMA_F32_16X16X4_F32 over K/4 steps.
//
// A operand (16x4 f32, 2 VGPRs/lane): lane L holds row m = L%16,
//   a[v] = A[m][kk + v + 2*(L/16)]  -> contiguous k-pair, loaded from LDS.
// B operand (4x16 f32, 2 VGPRs/lane): lane L holds col n = L%16,
//   b[v] = W[kk + v + 2*(L/16)][n].
// C/D (16x16 f32, 8 VGPRs/lane): acc[r] = element (m = r + 8*(L/16), n = L%16).

template <int K, bool RELU>
__global__ __launch_bounds__(256) void gemm_wmma_kernel(
    const float* __restrict__ A, const float* __restrict__ W,
    const float* __restrict__ bias, float* __restrict__ OUT) {
    __shared__ float As[16 * K];

    const int row0 = blockIdx.x * 16;

    // Cooperative, coalesced load of the 16 x K A-tile into LDS.
    for (int p = threadIdx.x; p < 16 * K; p += 256) {
        int r = p / K;
        int k = p % K;
        As[p] = A[(long long)(row0 + r) * K + k];
    }
    __syncthreads();

    const int wave = threadIdx.x >> 5;
    const int lane = threadIdx.x & 31;
    const int half = lane >> 4;   // 0 or 1
    const int l16  = lane & 15;   // row (A) / col (B,C,D) index

    const int n0 = (wave * 2) * 16;
    const int n1 = n0 + 16;

    v8f acc0 = {};
    v8f acc1 = {};

    for (int kk = 0; kk < K; kk += 4) {
        const int ka = kk + 2 * half;
        v2f a;
        a.x = As[l16 * K + ka];
        a.y = As[l16 * K + ka + 1];
        v2f b0;
        b0.x = W[(long long)ka * D_HID + n0 + l16];
        b0.y = W[(long long)(ka + 1) * D_HID + n0 + l16];
        v2f b1;
        b1.x = W[(long long)ka * D_HID + n1 + l16];
        b1.y = W[(long long)(ka + 1) * D_HID + n1 + l16];
        acc0 = __builtin_amdgcn_wmma_f32_16x16x4_f32(
            false, a, false, b0, (short)0, acc0, false, false);
        acc1 = __builtin_amdgcn_wmma_f32_16x16x4_f32(
            false, a, false, b1, (short)0, acc1, false, false);
    }

    const float bb0 = bias[n0 + l16];
    const float bb1 = bias[n1 + l16];
#pragma unroll
    for (int r = 0; r < 8; ++r) {
        const int m = row0 + r + 8 * half;
        float v0 = acc0[r] + bb0;
        float v1 = acc1[r] + bb1;
        if (RELU) {
            v0 = v0 > 0.0f ? v0 : 0.0f;
            v1 = v1 > 0.0f ? v1 : 0.0f;
        }
        OUT[(long long)m * D_HID + n0 + l16] = v0;
        OUT[(long long)m * D_HID + n1 + l16] = v1;
    }
}

// ---------------- mean-pool accumulation over batch ids ----------------

__global__ void pool_add_kernel(const float* __restrict__ h,
                                const int* __restrict__ batch,
                                float* __restrict__ pooled,
                                float* __restrict__ counts) {
    long long idx = (long long)blockIdx.x * blockDim.x + threadIdx.x;
    long long tot = (long long)N_NODES * (D_HID / 4);
    if (idx >= tot) return;
    int node = (int)(idx >> 6);   // / 64 groups
    int j = (int)(idx & 63);
    int b = batch[node];
    float4 v = *(const float4*)(h + (long long)node * D_HID + 4 * j);
    float* o = pooled + (long long)b * D_HID + 4 * j;
    atomicAdd(o + 0, v.x);
    atomicAdd(o + 1, v.y);
    atomicAdd(o + 2, v.z);
    atomicAdd(o + 3, v.w);
    if (j == 0) atomicAdd(&counts[b], 1.0f);
}

// ---------------- final: out[g,o] = (pooled[g]/max(count,1)) @ Wl + bl ------

__global__ void final_linear_kernel(const float* __restrict__ pooled,
                                    const float* __restrict__ counts,
                                    const float* __restrict__ Wl,
                                    const float* __restrict__ bl,
                                    float* __restrict__ out) {
    int t = threadIdx.x;  // 0..127
    if (t >= N_GRAPHS * D_OUT) return;
    int g = t >> 1;
    int o = t & 1;
    float c = counts[g];
    if (c < 1.0f) c = 1.0f;
    float s = 0.0f;
    for (int k = 0; k < D_HID; ++k)
        s += (pooled[g * D_HID + k] / c) * Wl[k * D_OUT + o];
    out[g * D_OUT + o] = s + bl[o];
}

// ---------------- launch ----------------

extern "C" void kernel_launch(void* const* d_in, const int* in_sizes, int n_in,
                              void* d_out, int out_size, void* d_ws, size_t ws_size,
                              hipStream_t stream) {
    const float* x  = (const float*)d_in[0];
    const int*   ei = (const int*)d_in[1];     // [2, N_EDGES]
    const int*   src = ei;
    const int*   dst = ei + N_EDGES;
    const int*   batch = (const int*)d_in[2];
    const float* W1 = (const float*)d_in[3];
    const float* b1 = (const float*)d_in[4];
    const float* W2 = (const float*)d_in[5];
    const float* b2 = (const float*)d_in[6];
    const float* V1 = (const float*)d_in[7];
    const float* c1 = (const float*)d_in[8];
    const float* V2 = (const float*)d_in[9];
    const float* c2 = (const float*)d_in[10];
    const float* Wl = (const float*)d_in[11];
    const float* bl = (const float*)d_in[12];
    float* out = (float*)d_out;

    // Workspace layout (bytes, 256-aligned):
    //   A: [N,96]  x + agg1                          19,200,000
    //   B: [N,256] t1 / t2                            51,200,000
    //   C: [N,256] h1 / h2                            51,200,000
    //   D: [N,256] h1 + agg2                          51,200,000
    //   pooled[64*256] + counts[64] contiguous
    char* ws = (char*)d_ws;
    float* A = (float*)(ws + 0);
    float* B = (float*)(ws + 19200000);
    float* C = (float*)(ws + 70400000);
    float* D = (float*)(ws + 121600000);
    float* pooled = (float*)(ws + 172800000);
    float* counts = pooled + N_GRAPHS * D_HID;

    const int T = 256;
    const int nMtiles = N_NODES / 16;  // 3125

    // ---- layer 1: A = x + segment_sum(x[src], dst) ----
    {
        int n4 = N_NODES * D_IN / 4;
        copy_f4_kernel<<<(n4 + T - 1) / T, T, 0, stream>>>(
            (const float4*)x, (float4*)A, n4);
        long long tot = (long long)N_EDGES * (D_IN / 4);
        scatter_add_kernel<<<(int)((tot + T - 1) / T), T, 0, stream>>>(
            x, src, dst, A, D_IN / 4, D_IN);
    }
    // B = relu(A @ W1 + b1)
    gemm_wmma_kernel<D_IN, true><<<nMtiles, T, 0, stream>>>(A, W1, b1, B);
    // C = relu(B @ W2 + b2)   (GIN output + outer relu)
    gemm_wmma_kernel<D_HID, true><<<nMtiles, T, 0, stream>>>(B, W2, b2, C);

    // ---- layer 2: D = C + segment_sum(C[src], dst) ----
    {
        int n4 = N_NODES * D_HID / 4;
        copy_f4_kernel<<<(n4 + T - 1) / T, T, 0, stream>>>(
            (const float4*)C, (float4*)D, n4);
        long long tot = (long long)N_EDGES * (D_HID / 4);
        scatter_add_kernel<<<(int)((tot + T - 1) / T), T, 0, stream>>>(
            C, src, dst, D, D_HID / 4, D_HID);
    }
    // B = relu(D @ V1 + c1)
    gemm_wmma_kernel<D_HID, true><<<nMtiles, T, 0, stream>>>(D, V1, c1, B);
    // C = B @ V2 + c2  (no relu)
    gemm_wmma_kernel<D_HID, false><<<nMtiles, T, 0, stream>>>(B, V2, c2, C);

    // ---- mean pool + head ----
    {
        int nz = N_GRAPHS * D_HID + N_GRAPHS;
        zero_f_kernel<<<(nz + T - 1) / T, T, 0, stream>>>(pooled, nz);
        long long tot = (long long)N_NODES * (D_HID / 4);
        pool_add_kernel<<<(int)((tot + T - 1) / T), T, 0, stream>>>(
            C, batch, pooled, counts);
        final_linear_kernel<<<1, 128, 0, stream>>>(pooled, counts, Wl, bl, out);
    }
}